// Att_multiply_42984032698905
// MI455X (gfx1250) — compile-verified
//
#include <hip/hip_runtime.h>
#include <math.h>

typedef __attribute__((ext_vector_type(2))) float v2f;
typedef __attribute__((ext_vector_type(4))) float v4f;
typedef __attribute__((ext_vector_type(8))) float v8f;

#define BATCH 32
#define NN    128
#define DD    128
#define NEG_SLOPE 0.01f

// ---------------------------------------------------------------------------
// Kernel 1: scores = (E .* w) @ E^T  (per batch, via V_WMMA_F32_16X16X4_F32),
// then bias + LeakyReLU + row softmax -> alphas [B,N,N].
// One block = 256 threads = 8 waves; block covers a 16-row strip of one batch.
// Wave w computes the 16x16 tile at columns [16w, 16w+16).
// ---------------------------------------------------------------------------
__global__ void __launch_bounds__(256)
scores_softmax_kernel(const float* __restrict__ emb,
                      const float* __restrict__ att_w,
                      const float* __restrict__ att_b,
                      float* __restrict__ alphas) {
  __shared__ float s_scores[16 * NN];  // 8 KB: 16 rows x 128 cols

  const int b   = blockIdx.x >> 3;         // NN/16 == 8 row-tiles per batch
  const int i0  = (blockIdx.x & 7) << 4;   // first row of this strip
  const int tid = threadIdx.x;
  const int wv  = tid >> 5;                // wave id 0..7 -> column tile
  const int ln  = tid & 31;                // lane within wave32
  const int j0  = wv << 4;

  const int m  = ln & 15;   // row-in-tile for A, col-in-tile source row for B
  const int kh = ln >> 4;   // K-half selector (K = 2*kh + {0,1})

  const float* eb   = emb + (size_t)b * NN * DD;
  const float* arow = eb + (size_t)(i0 + m) * DD;  // A source: row i0+m
  const float* brow = eb + (size_t)(j0 + m) * DD;  // B source: row j0+m

  v8f acc = {};
#pragma unroll
  for (int k0 = 0; k0 < DD; k0 += 4) {
    const int d = k0 + 2 * kh;               // even -> 8B aligned float2 loads
    v2f ea = *(const v2f*)(arow + d);
    v2f wb = *(const v2f*)(att_w + d);
    v2f bb = *(const v2f*)(brow + d);
    v2f a;
    a.x = ea.x * wb.x;                       // A = E .* w  (column scaling)
    a.y = ea.y * wb.y;
    // D = A(16x4) x B(4x16) + C ; f32 in / f32 out
    acc = __builtin_amdgcn_wmma_f32_16x16x4_f32(
        /*neg_a=*/false, a, /*neg_b=*/false, bb,
        /*c_mod=*/(short)0, acc, /*reuse_a=*/false, /*reuse_b=*/false);
  }

  // Scatter accumulator to LDS per ISA C/D layout:
  //   lane L, vgpr r  ->  row = r + 8*(L/16), col = L%16
  const float bias = att_b[0];
#pragma unroll
  for (int r = 0; r < 8; ++r) {
    float s = acc[r] + bias;
    s = (s >= 0.0f) ? s : NEG_SLOPE * s;     // LeakyReLU
    const int row = r + 8 * kh;
    s_scores[row * NN + (j0 + m)] = s;
  }
  __syncthreads();

  // Row softmax: 16 rows, one thread per row (128-wide serial scan; this
  // kernel is tiny next to the 256MB value write, simplicity wins).
  if (tid < 16) {
    const int row = tid;
    float* srow = &s_scores[row * NN];
    float mx = -INFINITY;
    for (int j = 0; j < NN; ++j) mx = fmaxf(mx, srow[j]);
    float sum = 0.0f;
    for (int j = 0; j < NN; ++j) {
      const float ex = __expf(srow[j] - mx);
      srow[j] = ex;
      sum += ex;
    }
    const float inv = 1.0f / sum;
    float* out = alphas + ((size_t)(b * NN) + (i0 + row)) * NN;
    for (int j = 0; j < NN; ++j) out[j] = srow[j] * inv;
  }
}

// ---------------------------------------------------------------------------
// Kernel 2: value[b,i,j,:] = e[b,i,:] * e[b,j,:]   (256 MB, pure BW).
// One block per (b,i). Thread owns one float4 chunk of D (32 chunks),
// 8 waves stripe over j. Non-temporal stores: output >> L2, consumed by host.
// ---------------------------------------------------------------------------
__global__ void __launch_bounds__(256)
value_kernel(const float* __restrict__ emb, float* __restrict__ value) {
  const int b   = blockIdx.x >> 7;   // /NN
  const int i   = blockIdx.x & (NN - 1);
  const int tid = threadIdx.x;
  const int c   = tid & 31;          // float4 chunk index: d = 4*c
  const int jg  = tid >> 5;          // 0..7

  const v4f* erow_i = (const v4f*)(emb + ((size_t)(b * NN + i)) * DD);
  const v4f ei = erow_i[c];

  v4f* out = (v4f*)(value + (((size_t)(b * NN) + i) * NN) * DD);

  for (int j = jg; j < NN; j += 8) {
    const v4f* erow_j = (const v4f*)(emb + ((size_t)(b * NN + j)) * DD);
    const v4f ej = erow_j[c];
    v4f v;
    v.x = ei.x * ej.x;
    v.y = ei.y * ej.y;
    v.z = ei.z * ej.z;
    v.w = ei.w * ej.w;
    __builtin_nontemporal_store(v, out + (size_t)j * (DD / 4) + c);
  }
}

// ---------------------------------------------------------------------------
extern "C" void kernel_launch(void* const* d_in, const int* in_sizes, int n_in,
                              void* d_out, int out_size, void* d_ws, size_t ws_size,
                              hipStream_t stream) {
  const float* emb   = (const float*)d_in[0];  // [B,N,D]
  const float* att_w = (const float*)d_in[1];  // [D]
  const float* att_b = (const float*)d_in[2];  // [1]

  float* alphas = (float*)d_out;                          // [B,N,N]
  float* value  = (float*)d_out + (size_t)BATCH * NN * NN; // [B,N,N,D]

  // Kernel 1: 32 batches * 8 row-tiles = 256 blocks of 256 threads.
  scores_softmax_kernel<<<BATCH * (NN / 16), 256, 0, stream>>>(
      emb, att_w, att_b, alphas);

  // Kernel 2: one block per (b,i) = 4096 blocks of 256 threads.
  value_kernel<<<BATCH * NN, 256, 0, stream>>>(emb, value);
}